// holt_winters_no_trend_25692494365657
// MI455X (gfx1250) — compile-verified
//
#include <hip/hip_runtime.h>
#include <hip/hip_bf16.h>

// Holt-Winters (no trend) collapsed to GEMM:
//   out[b, j] = sum_t W[t,j] * series[b,t] + V[shift[b], j]
// W: 2048 x 24 (padded to 32 cols), V: 24 x 24, both depend only on
// sigmoid(alpha), sigmoid(gamma), init_season -> built once by one wave.
// Bulk work: [32768 x 2048] @ [2048 x 32] fp32 GEMM via v_wmma_f32_16x16x4_f32.
// Memory-bound: 256 MB series stream -> ~11 us at 23.3 TB/s.

#define HW_B     32768
#define HW_T     2048
#define HW_SLEN  24
#define HW_NPRED 24
#define HW_NPAD  32            // W padded to 32 columns (cols 24..31 == 0)
#define HW_KCH   128           // K-chunk staged in LDS
#define HW_NCH   (HW_T / HW_KCH)
#define HW_LDST  40            // LDS row stride in floats (bank-conflict free)

typedef __attribute__((ext_vector_type(2))) float v2f;
typedef __attribute__((ext_vector_type(8))) float v8f;
typedef __attribute__((ext_vector_type(4))) int   v4i;

#if defined(__HIP_DEVICE_COMPILE__) && __has_builtin(__builtin_amdgcn_wmma_f32_16x16x4_f32)
#define HW_USE_WMMA 1
#else
#define HW_USE_WMMA 0
#endif

#if defined(__HIP_DEVICE_COMPILE__) && \
    __has_builtin(__builtin_amdgcn_global_load_async_to_lds_b128) && \
    __has_builtin(__builtin_amdgcn_s_wait_asynccnt)
#define HW_USE_ASYNC 1
#else
#define HW_USE_ASYNC 0
#endif

// ---------------------------------------------------------------------------
// Phase A: adjoint (reverse-mode) sweep of the linear recurrence.
// One wave; lane j (j<24) owns output j; lanes 24..31 produce the zero pad.
// Forward step t (idx = t%24):
//   ns      = (1-a)*smooth - a*seas[idx] + a*x_t
//   nseas_i = -g(1-a)*smooth + (g*a+1-g)*seas[idx] + g(1-a)*x_t
// Adjoint (ws = d out_j / d smooth, wseas = d out_j / d seas):
//   W[t,j]     = ws*a + wi*g(1-a)
//   ws'        = ws*(1-a) - wi*g(1-a)
//   wseas[idx] = wi*(g*a+1-g) - ws*a
// Init at t = T-1 end state: ws = 1, wseas[(T+j)%24] = 1.
// ---------------------------------------------------------------------------
__global__ void hw_build_weights_kernel(const float* __restrict__ alpha,
                                        const float* __restrict__ gamma,
                                        const float* __restrict__ init_season,
                                        float* __restrict__ W,   // [HW_T][HW_NPAD]
                                        float* __restrict__ V) { // [HW_SLEN][HW_NPRED]
  const int j = threadIdx.x;  // 0..31
  const float a    = 1.0f / (1.0f + expf(-alpha[0]));
  const float g    = 1.0f / (1.0f + expf(-gamma[0]));
  const float oma  = 1.0f - a;
  const float g1a  = g * oma;
  const float keep = g * a + 1.0f - g;

  float ws = (j < HW_NPRED) ? 1.0f : 0.0f;
  float wseas[HW_SLEN];
  const int p = (HW_T + j) % HW_SLEN;   // pred_idx[j] = (2048+j)%24
#pragma unroll
  for (int k = 0; k < HW_SLEN; ++k)
    wseas[k] = (j < HW_NPRED && k == p) ? 1.0f : 0.0f;

  int t = HW_T - 1;  // 2047, idx sequence 7,6,...,1, then (0,23,...,1) x 85
  auto step = [&](int idx) {
    const float wi   = wseas[idx];
    const float wcol = ws * a + wi * g1a;
    const float nws  = ws * oma - wi * g1a;
    wseas[idx] = wi * keep - ws * a;
    ws = nws;
    W[t * HW_NPAD + j] = wcol;
    --t;
  };

#pragma unroll
  for (int u = 0; u < 7; ++u) step(7 - u);          // t = 2047..2041
  for (int blk = 0; blk < 85; ++blk) {              // t = 2040..1
#pragma unroll
    for (int u = 0; u < 24; ++u) step((24 - u) % 24);
  }
  W[j] = ws;  // t == 0 column: coefficient of series[:,0] (initial smooth)

  // Per-shift seasonal contribution: V[s][j] = sum_k wseas[k]*init_season[(k-s)%24]
  if (j < HW_NPRED) {
    for (int s = 0; s < HW_SLEN; ++s) {
      float acc = 0.0f;
#pragma unroll
      for (int k = 0; k < HW_SLEN; ++k)
        acc += wseas[k] * init_season[(k - s + HW_SLEN) % HW_SLEN];
      V[s * HW_NPRED + j] = acc;
    }
  }
}

// ---------------------------------------------------------------------------
// LDS staging of one W chunk (128 rows x 32 floats, LDS row stride 40).
// Async path uses gfx1250 GLOBAL_LOAD_ASYNC_TO_LDS_B128 (ASYNCcnt-tracked).
// ---------------------------------------------------------------------------
__device__ __forceinline__ void hw_stage_chunk(float* ldsbuf,
                                               const float* gsrc,
                                               int tid) {
#pragma unroll
  for (int i = 0; i < 4; ++i) {
    const int pc  = tid + i * 256;   // 1024 float4 pieces
    const int row = pc >> 3;
    const int q   = pc & 7;
    float4* d = (float4*)(ldsbuf + row * HW_LDST + q * 4);
    const float4* s = (const float4*)(gsrc + row * HW_NPAD + q * 4);
#if HW_USE_ASYNC
    __builtin_amdgcn_global_load_async_to_lds_b128(
        (__attribute__((address_space(1))) v4i*)s,
        (__attribute__((address_space(3))) v4i*)d, 0, 0);
#else
    *d = *s;
#endif
  }
}

__device__ __forceinline__ void hw_wait_async() {
#if HW_USE_ASYNC
  __builtin_amdgcn_s_wait_asynccnt(0);
#endif
}

// ---------------------------------------------------------------------------
// Phase B: GEMM. 8 waves/block, each wave computes a 16-row x 32-col tile
// with v_wmma_f32_16x16x4_f32 (two N-tiles of 16). Double-buffered W in LDS.
// A-frag (16x4 f32): lane = 16*half + m holds A[m][2*half + {0,1}] -> float2.
// B-frag (4x16 f32): vgpr0 = row K=2*half, vgpr1 = row K=2*half+1, col = lane&15.
// C/D (16x16 f32): vgpr r, lane = 16*half + n -> D[r + 8*half][n].
// ---------------------------------------------------------------------------
__global__ __launch_bounds__(256) void hw_gemm_kernel(
    const float* __restrict__ series,   // [HW_B][HW_T]
    const int*   __restrict__ shifts,   // [HW_B]
    const float* __restrict__ W,        // [HW_T][HW_NPAD]
    const float* __restrict__ V,        // [HW_SLEN][HW_NPRED]
    float* __restrict__ out) {          // [HW_B][HW_NPRED]
  __shared__ float Wl[2][HW_KCH * HW_LDST];   // 2 x 20 KB

  const int tid  = threadIdx.x;
  const int lane = tid & 31;
  const int wave = tid >> 5;

#if HW_USE_WMMA
  const int half = lane >> 4;
  const int n    = lane & 15;
  const int row0 = blockIdx.x * 128 + wave * 16;
  const float* arow = series + (size_t)(row0 + n) * HW_T + 2 * half;
  v8f c0 = {0.f, 0.f, 0.f, 0.f, 0.f, 0.f, 0.f, 0.f};
  v8f c1 = {0.f, 0.f, 0.f, 0.f, 0.f, 0.f, 0.f, 0.f};
#else
  const int rowS = blockIdx.x * 128 + tid;    // threads 0..127 each own a row
  float acc[HW_NPRED];
#pragma unroll
  for (int jj = 0; jj < HW_NPRED; ++jj) acc[jj] = 0.0f;
#endif

  hw_stage_chunk(&Wl[0][0], W, tid);

  for (int c = 0; c < HW_NCH; ++c) {
    hw_wait_async();
    __syncthreads();
    if (c + 1 < HW_NCH)
      hw_stage_chunk(&Wl[(c + 1) & 1][0], W + (size_t)(c + 1) * HW_KCH * HW_NPAD, tid);

    const float* Wb = &Wl[c & 1][0];
#if HW_USE_WMMA
#pragma unroll 4
    for (int k = 0; k < HW_KCH; k += 4) {
      const float2 av = *(const float2*)(arow + c * HW_KCH + k);
      v2f afrag; afrag.x = av.x; afrag.y = av.y;
      v2f b0, b1;
      const int r0i = (k + 2 * half) * HW_LDST;
      b0.x = Wb[r0i + n];
      b0.y = Wb[r0i + HW_LDST + n];
      b1.x = Wb[r0i + 16 + n];
      b1.y = Wb[r0i + HW_LDST + 16 + n];
      c0 = __builtin_amdgcn_wmma_f32_16x16x4_f32(false, afrag, false, b0,
                                                 (short)0, c0, false, false);
      c1 = __builtin_amdgcn_wmma_f32_16x16x4_f32(false, afrag, false, b1,
                                                 (short)0, c1, false, false);
    }
#else
    if (tid < 128) {
      const float* srow = series + (size_t)rowS * HW_T + c * HW_KCH;
      for (int kk = 0; kk < HW_KCH; ++kk) {
        const float x = srow[kk];
#pragma unroll
        for (int jj = 0; jj < HW_NPRED; ++jj)
          acc[jj] += x * Wb[kk * HW_LDST + jj];
      }
    }
#endif
  }

#if HW_USE_WMMA
#pragma unroll
  for (int r = 0; r < 8; ++r) {
    const int row = row0 + r + 8 * half;
    const int sh  = shifts[row];
    out[row * HW_NPRED + n] = c0[r] + V[sh * HW_NPRED + n];
    if (n < 8)
      out[row * HW_NPRED + 16 + n] = c1[r] + V[sh * HW_NPRED + 16 + n];
  }
#else
  if (tid < 128) {
    const int sh = shifts[rowS];
#pragma unroll
    for (int jj = 0; jj < HW_NPRED; ++jj)
      out[rowS * HW_NPRED + jj] = acc[jj] + V[sh * HW_NPRED + jj];
  }
#endif
}

// ---------------------------------------------------------------------------
extern "C" void kernel_launch(void* const* d_in, const int* in_sizes, int n_in,
                              void* d_out, int out_size, void* d_ws, size_t ws_size,
                              hipStream_t stream) {
  const float* series      = (const float*)d_in[0];
  const int*   shifts      = (const int*)d_in[1];
  const float* alpha       = (const float*)d_in[2];
  const float* gamma       = (const float*)d_in[3];
  const float* init_season = (const float*)d_in[4];
  // d_in[5] = n_preds (compile-time 24 here)

  float* W = (float*)d_ws;                    // 2048*32 floats = 256 KB
  float* V = W + HW_T * HW_NPAD;              // + 24*24 floats
  float* out = (float*)d_out;

  hw_build_weights_kernel<<<dim3(1), dim3(32), 0, stream>>>(
      alpha, gamma, init_season, W, V);
  hw_gemm_kernel<<<dim3(HW_B / 128), dim3(256), 0, stream>>>(
      series, shifts, W, V, out);
}